// GNNGraphExtractor_17154099380300
// MI455X (gfx1250) — compile-verified
//
#include <hip/hip_runtime.h>
#include <math.h>

#define B_  16
#define N_  2048
#define E_  65536
#define D_  128
#define H_  256
#define L_  3
#define NEG_SLOPE 0.01f
#define BN_EPS 1e-5f

#define BN_ROWS (B_ * N_)     // 32768 aggregate rows

typedef __attribute__((ext_vector_type(2))) float v2f;
typedef __attribute__((ext_vector_type(8))) float v8f;

// ---------------------------------------------------------------------------
// Zero-fill (u32 granularity)
// ---------------------------------------------------------------------------
__global__ __launch_bounds__(256) void zero_u32(unsigned* __restrict__ p, int n) {
    int i = blockIdx.x * 256 + threadIdx.x;
    if (i < n) p[i] = 0u;
}

// ---------------------------------------------------------------------------
// f32 WMMA GEMM:  Y[M x 256] = A1[M x K] @ W1[256 x K]^T (+ A2 @ W2^T) + bias
// Block = 256 threads = 8 waves; block tile 256x64; wave tile 32x64
// (8 accumulators of V_WMMA_F32_16X16X4_F32, 2 A fragments / k-step).
// B operand staged in LDS: 64 x K panel, row stride K+4 floats -> fragment
// reads hit banks (4*row + k + 2*lh) mod 64, conflict-free for all 32 lanes,
// and addresses are base + constant -> DS immediate offsets (no VALU math).
// ---------------------------------------------------------------------------

template<int K>
__device__ __forceinline__ void stage_w(const float* __restrict__ W,
                                        int col0, float* __restrict__ ws,
                                        int tid) {
    constexpr int LDK = K + 4;
    constexpr int K4  = K >> 2;               // float4s per row
    constexpr int ITERS = (64 * K4) >> 8;     // / 256 threads
#pragma unroll
    for (int it = 0; it < ITERS; ++it) {
        int idx = it * 256 + tid;             // float4 index
        int row = idx / K4;
        int kq  = (idx - row * K4) * 4;
        float4 v = *(const float4*)(W + (size_t)(col0 + row) * K + kq);
        *(float4*)(ws + row * LDK + kq) = v;
    }
}

template<int K>
__device__ __forceinline__ void mma_panel(const float* __restrict__ A,
                                          const float* __restrict__ ws,
                                          int row0, int lr, int lh,
                                          v8f acc[8]) {
    constexpr int LDK = K + 4;
    const float* a0 = A + (size_t)(row0 + lr) * K + lh * 2;        // rows 0..15
    const float* a1 = A + (size_t)(row0 + 16 + lr) * K + lh * 2;   // rows 16..31
    const float* b0 = ws + (0 * 16 + lr) * LDK + lh * 2;
    const float* b1 = ws + (1 * 16 + lr) * LDK + lh * 2;
    const float* b2 = ws + (2 * 16 + lr) * LDK + lh * 2;
    const float* b3 = ws + (3 * 16 + lr) * LDK + lh * 2;

#pragma unroll 8
    for (int k = 0; k < K; k += 4) {
        v2f av0 = *(const v2f*)(a0 + k);
        v2f av1 = *(const v2f*)(a1 + k);
        v2f bv0 = *(const v2f*)(b0 + k);
        v2f bv1 = *(const v2f*)(b1 + k);
        v2f bv2 = *(const v2f*)(b2 + k);
        v2f bv3 = *(const v2f*)(b3 + k);
        acc[0] = __builtin_amdgcn_wmma_f32_16x16x4_f32(false, av0, false, bv0, (short)0, acc[0], false, false);
        acc[1] = __builtin_amdgcn_wmma_f32_16x16x4_f32(false, av0, false, bv1, (short)0, acc[1], false, false);
        acc[2] = __builtin_amdgcn_wmma_f32_16x16x4_f32(false, av0, false, bv2, (short)0, acc[2], false, false);
        acc[3] = __builtin_amdgcn_wmma_f32_16x16x4_f32(false, av0, false, bv3, (short)0, acc[3], false, false);
        acc[4] = __builtin_amdgcn_wmma_f32_16x16x4_f32(false, av1, false, bv0, (short)0, acc[4], false, false);
        acc[5] = __builtin_amdgcn_wmma_f32_16x16x4_f32(false, av1, false, bv1, (short)0, acc[5], false, false);
        acc[6] = __builtin_amdgcn_wmma_f32_16x16x4_f32(false, av1, false, bv2, (short)0, acc[6], false, false);
        acc[7] = __builtin_amdgcn_wmma_f32_16x16x4_f32(false, av1, false, bv3, (short)0, acc[7], false, false);
    }
}

template<int K, int HAS_A2>
__global__ __launch_bounds__(256) void wmma_gemm(
    const float* __restrict__ A1, const float* __restrict__ W1,
    const float* __restrict__ A2, const float* __restrict__ W2,
    const float* __restrict__ bias, float* __restrict__ Y)
{
    __shared__ float ws[64 * (K + 4)];        // 66.5KB (K=256) / 33.8KB (K=128)

    const int tid  = threadIdx.x;
    const int lane = tid & 31;
    const int wave = tid >> 5;
    const int lr   = lane & 15;
    const int lh   = lane >> 4;
    const int row0 = blockIdx.x * 256 + wave * 32;
    const int col0 = blockIdx.y * 64;

    // Pull the second A panel toward the caches while panel 1 computes.
    if (HAS_A2) {
        const float* p = A2 + (size_t)(row0 + lr) * K;
        __builtin_prefetch(p + lh * 32, 0, 3);
        __builtin_prefetch(p + 64 + lh * 32, 0, 3);
        const float* q = A2 + (size_t)(row0 + 16 + lr) * K;
        __builtin_prefetch(q + lh * 32, 0, 3);
        __builtin_prefetch(q + 64 + lh * 32, 0, 3);
    }

    v8f acc[8] = {v8f{}, v8f{}, v8f{}, v8f{}, v8f{}, v8f{}, v8f{}, v8f{}};

    stage_w<K>(W1, col0, ws, tid);
    __syncthreads();
    mma_panel<K>(A1, ws, row0, lr, lh, acc);

    if (HAS_A2) {
        __syncthreads();                      // everyone done reading panel 1
        stage_w<K>(W2, col0, ws, tid);
        __syncthreads();
        mma_panel<K>(A2, ws, row0, lr, lh, acc);
    }

    // C/D layout: vgpr i of acc[rh*4+j] -> row row0 + rh*16 + lh*8 + i,
    // column col0 + j*16 + (lane&15)
#pragma unroll
    for (int rh = 0; rh < 2; ++rh) {
        const int rbase = row0 + rh * 16 + lh * 8;
#pragma unroll
        for (int j = 0; j < 4; ++j) {
            const int c  = col0 + j * 16 + lr;
            const float bv = bias[c];
            const v8f av = acc[rh * 4 + j];
#pragma unroll
            for (int i = 0; i < 8; ++i) {
                Y[(size_t)(rbase + i) * H_ + c] = av[i] + bv;
            }
        }
    }
}

// ---------------------------------------------------------------------------
// CSR build (edges grouped by destination). Done ONCE: the graph is constant
// across all 3 conv layers, so 805M f32 scatter-atomics become ~2M int
// atomics + 3 atomic-free gather passes.
// ---------------------------------------------------------------------------
__global__ __launch_bounds__(256) void count_deg(const int* __restrict__ edge_index,
                                                 int* __restrict__ deg)
{
    const int idx = blockIdx.x * 256 + threadIdx.x;   // 0 .. B*E
    const int b = idx >> 16;                          // E_ = 65536
    const int e = idx & (E_ - 1);
    const int d = edge_index[(size_t)b * 2 * E_ + E_ + e];
    atomicAdd(&deg[b * N_ + d], 1);
}

// Single block: exclusive prefix sum over BN_ROWS degree counts.
// 256 threads x 128 serial elements + Hillis-Steele over 256 partials.
__global__ __launch_bounds__(256) void scan_rowptr(const int* __restrict__ deg,
                                                   int* __restrict__ rowptr)
{
    __shared__ int part[256];
    const int t = threadIdx.x;
    const int base = t * (BN_ROWS / 256);             // 128 per thread
    int s = 0;
    for (int j = 0; j < BN_ROWS / 256; ++j) s += deg[base + j];
    part[t] = s;
    __syncthreads();
#pragma unroll
    for (int off = 1; off < 256; off <<= 1) {
        int v = (t >= off) ? part[t - off] : 0;
        __syncthreads();
        part[t] += v;
        __syncthreads();
    }
    int run = (t == 0) ? 0 : part[t - 1];
    for (int j = 0; j < BN_ROWS / 256; ++j) {
        rowptr[base + j] = run;
        run += deg[base + j];
    }
}

__global__ __launch_bounds__(256) void fill_perm(const int* __restrict__ edge_index,
                                                 const int* __restrict__ rowptr,
                                                 int* __restrict__ cursor,
                                                 int* __restrict__ perm)
{
    const int idx = blockIdx.x * 256 + threadIdx.x;   // 0 .. B*E
    const int b = idx >> 16;
    const int e = idx & (E_ - 1);
    const int bd = b * N_ + edge_index[(size_t)b * 2 * E_ + E_ + e];
    const int pos = atomicAdd(&cursor[bd], 1);
    perm[rowptr[bd] + pos] = e;                       // batch-local edge id
}

// ---------------------------------------------------------------------------
// Atomic-free aggregation: one wave per destination row (b,d).
// Lanes cooperatively load 32 edges' (src, weight), broadcast via shfl,
// each lane accumulates its 8 channels of h[src] (coalesced 1KB row reads,
// L2-resident). Row written exactly once (degree-0 rows write zeros, so no
// separate zero-fill pass is needed).
// ---------------------------------------------------------------------------
__global__ __launch_bounds__(256) void gather_rows(
    const float* __restrict__ h, const int* __restrict__ edge_index,
    const float* __restrict__ ew, const int* __restrict__ rowptr,
    const int* __restrict__ deg, const int* __restrict__ perm,
    float* __restrict__ agg)
{
    const int lane = threadIdx.x & 31;
    const int wave = threadIdx.x >> 5;
    const int bd   = blockIdx.x * 8 + wave;           // 0 .. BN_ROWS
    const int b    = bd >> 11;                        // N_ = 2048
    const int start = rowptr[bd];
    const int cnt   = deg[bd];

    const int*   srcp = edge_index + (size_t)b * 2 * E_;
    const float* ewp  = ew + (size_t)b * E_;
    const float* hb   = h + (size_t)b * N_ * H_;
    const int cbase = lane * 8;

    float4 acc0 = {0,0,0,0}, acc1 = {0,0,0,0};

    for (int basei = 0; basei < cnt; basei += 32) {
        const int myi = basei + lane;
        int   s = 0;
        float w = 0.f;
        if (myi < cnt) {
            const int e = perm[start + myi];
            s = srcp[e];
            w = ewp[e];
        }
        const int lim = min(32, cnt - basei);
        for (int j = 0; j < lim; ++j) {
            const int   sj = __shfl(s, j, 32);
            const float wj = __shfl(w, j, 32);
            const float4* hp = (const float4*)(hb + (size_t)sj * H_ + cbase);
            float4 v0 = hp[0];
            float4 v1 = hp[1];
            acc0.x = fmaf(v0.x, wj, acc0.x); acc0.y = fmaf(v0.y, wj, acc0.y);
            acc0.z = fmaf(v0.z, wj, acc0.z); acc0.w = fmaf(v0.w, wj, acc0.w);
            acc1.x = fmaf(v1.x, wj, acc1.x); acc1.y = fmaf(v1.y, wj, acc1.y);
            acc1.z = fmaf(v1.z, wj, acc1.z); acc1.w = fmaf(v1.w, wj, acc1.w);
        }
    }

    float4* ap = (float4*)(agg + (size_t)bd * H_ + cbase);
    ap[0] = acc0;
    ap[1] = acc1;
}

// ---------------------------------------------------------------------------
// BatchNorm over flat (B*N, H): stage 1 partial sums, stage 2 finalize,
// stage 3 apply + leaky relu (in place).
// stats layout: [0:H)=sum, [H:2H)=sumsq, [2H:3H)=scale, [3H:4H)=shift
// ---------------------------------------------------------------------------
__global__ __launch_bounds__(256) void bn_stats(const float* __restrict__ h,
                                                float* __restrict__ stats,
                                                int rowsPerBlock)
{
    const int c  = threadIdx.x;           // channel (H_ == blockDim)
    const int r0 = blockIdx.x * rowsPerBlock;
    float s = 0.f, ss = 0.f;
    for (int r = r0; r < r0 + rowsPerBlock; ++r) {
        float v = h[(size_t)r * H_ + c];
        s  += v;
        ss += v * v;
    }
    unsafeAtomicAdd(&stats[c],      s);
    unsafeAtomicAdd(&stats[H_ + c], ss);
}

__global__ __launch_bounds__(256) void bn_finalize(float* __restrict__ stats,
                                                   const float* __restrict__ gamma,
                                                   const float* __restrict__ beta)
{
    const int c = threadIdx.x;
    const float inv_cnt = 1.0f / (float)(B_ * N_);
    float mean = stats[c] * inv_cnt;
    float var  = stats[H_ + c] * inv_cnt - mean * mean;
    float sc   = gamma[c] * rsqrtf(var + BN_EPS);
    stats[2 * H_ + c] = sc;
    stats[3 * H_ + c] = beta[c] - mean * sc;
}

__global__ __launch_bounds__(256) void bn_apply(float* __restrict__ h,
                                                const float* __restrict__ stats)
{
    const int c = threadIdx.x;            // channel (H_ == blockDim)
    const size_t i = (size_t)blockIdx.x * H_ + c;
    float v = fmaf(h[i], stats[2 * H_ + c], stats[3 * H_ + c]);
    h[i] = v > 0.f ? v : NEG_SLOPE * v;
}

// ---------------------------------------------------------------------------
// Fused pooling head, barrier-free main loop. One block per batch, 8 waves.
// Each wave owns rows n = wave, wave+8, ... Each lane owns 8 channels.
// Logits computed fully in-wave (shfl_xor butterfly -> all lanes), softmax
// (K=2), per-lane channel accumulators; single LDS merge at the end.
// ---------------------------------------------------------------------------
__global__ __launch_bounds__(256) void pool_head(
    const float* __restrict__ h, const float* __restrict__ pool_w,
    const float* __restrict__ pool_b, float* __restrict__ out)
{
    __shared__ float acc_s[2][8][H_];      // [k][wave][channel]
    __shared__ float red[8];

    const int b    = blockIdx.x;
    const int tid  = threadIdx.x;
    const int lane = tid & 31;
    const int wave = tid >> 5;
    const int cbase = lane * 8;

    float4 w0a = *(const float4*)(pool_w + cbase);
    float4 w0b = *(const float4*)(pool_w + cbase + 4);
    float4 w1a = *(const float4*)(pool_w + H_ + cbase);
    float4 w1b = *(const float4*)(pool_w + H_ + cbase + 4);
    const float pb0 = pool_b[0];
    const float pb1 = pool_b[1];

    const float* hb = h + (size_t)b * N_ * H_;

    float4 a0a = {0,0,0,0}, a0b = {0,0,0,0};
    float4 a1a = {0,0,0,0}, a1b = {0,0,0,0};

    for (int n = wave; n < N_; n += 8) {
        const float* hr = hb + (size_t)n * H_ + cbase;
        float4 va = *(const float4*)(hr);
        float4 vb = *(const float4*)(hr + 4);

        float p0 = va.x*w0a.x + va.y*w0a.y + va.z*w0a.z + va.w*w0a.w
                 + vb.x*w0b.x + vb.y*w0b.y + vb.z*w0b.z + vb.w*w0b.w;
        float p1 = va.x*w1a.x + va.y*w1a.y + va.z*w1a.z + va.w*w1a.w
                 + vb.x*w1b.x + vb.y*w1b.y + vb.z*w1b.z + vb.w*w1b.w;
#pragma unroll
        for (int m = 16; m > 0; m >>= 1) {
            p0 += __shfl_xor(p0, m, 32);
            p1 += __shfl_xor(p1, m, 32);
        }
        float l0 = p0 + pb0, l1 = p1 + pb1;
        float mx = fmaxf(l0, l1);
        float e0 = __expf(l0 - mx), e1 = __expf(l1 - mx);
        float inv = 1.0f / (e0 + e1);
        float s0 = e0 * inv, s1 = e1 * inv;

        a0a.x = fmaf(va.x, s0, a0a.x); a0a.y = fmaf(va.y, s0, a0a.y);
        a0a.z = fmaf(va.z, s0, a0a.z); a0a.w = fmaf(va.w, s0, a0a.w);
        a0b.x = fmaf(vb.x, s0, a0b.x); a0b.y = fmaf(vb.y, s0, a0b.y);
        a0b.z = fmaf(vb.z, s0, a0b.z); a0b.w = fmaf(vb.w, s0, a0b.w);
        a1a.x = fmaf(va.x, s1, a1a.x); a1a.y = fmaf(va.y, s1, a1a.y);
        a1a.z = fmaf(va.z, s1, a1a.z); a1a.w = fmaf(va.w, s1, a1a.w);
        a1b.x = fmaf(vb.x, s1, a1b.x); a1b.y = fmaf(vb.y, s1, a1b.y);
        a1b.z = fmaf(vb.z, s1, a1b.z); a1b.w = fmaf(vb.w, s1, a1b.w);
    }

    *(float4*)(&acc_s[0][wave][cbase])     = a0a;
    *(float4*)(&acc_s[0][wave][cbase + 4]) = a0b;
    *(float4*)(&acc_s[1][wave][cbase])     = a1a;
    *(float4*)(&acc_s[1][wave][cbase + 4]) = a1b;
    __syncthreads();

    // thread tid owns channel c = tid
    float acc0 = 0.f, acc1 = 0.f;
#pragma unroll
    for (int wv = 0; wv < 8; ++wv) {
        acc0 += acc_s[0][wv][tid];
        acc1 += acc_s[1][wv][tid];
    }
    float o0 = acc0 > 0.f ? acc0 : NEG_SLOPE * acc0;
    float o1 = acc1 > 0.f ? acc1 : NEG_SLOPE * acc1;
    float xp = 0.5f * (o0 + o1);

    float q = xp * xp;
#pragma unroll
    for (int m = 16; m > 0; m >>= 1) q += __shfl_xor(q, m, 32);
    if (lane == 0) red[wave] = q;
    __syncthreads();
    float tot = 0.f;
#pragma unroll
    for (int wv = 0; wv < 8; ++wv) tot += red[wv];
    float nrm = sqrtf(tot);
    out[(size_t)b * H_ + tid] = xp / fmaxf(nrm, 1e-12f);
}

// ---------------------------------------------------------------------------
// Host-side launch
// ---------------------------------------------------------------------------
extern "C" void kernel_launch(void* const* d_in, const int* in_sizes, int n_in,
                              void* d_out, int out_size, void* d_ws, size_t ws_size,
                              hipStream_t stream)
{
    const float* x            = (const float*)d_in[0];
    const int*   edge_index   = (const int*)  d_in[1];
    const float* edge_feature = (const float*)d_in[2];
    const float* pre_w        = (const float*)d_in[3];
    const float* pre_b        = (const float*)d_in[4];
    const float* rel_w        = (const float*)d_in[5];
    const float* rel_b        = (const float*)d_in[6];
    const float* root_w       = (const float*)d_in[7];
    const float* bn_gamma     = (const float*)d_in[8];
    const float* bn_beta      = (const float*)d_in[9];
    const float* pool_w       = (const float*)d_in[10];
    const float* pool_b       = (const float*)d_in[11];

    const size_t BNH = (size_t)B_ * N_ * H_;        // 8,388,608 floats
    float* hA    = (float*)d_ws;                    // current node features
    float* hB    = hA + BNH;                        // next node features
    float* agg   = hB + BNH;                        // gather accumulator
    float* stats = agg + BNH;                       // 4*H floats of BN state
    int* deg     = (int*)(stats + 4 * H_);          // BN_ROWS
    int* rowptr  = deg + BN_ROWS;                   // BN_ROWS
    int* cursor  = rowptr + BN_ROWS;                // BN_ROWS
    int* perm    = cursor + BN_ROWS;                // B*E

    const int M = B_ * N_;                          // 32768 rows
    const dim3 ggrid(M / 256, H_ / 64);

    // ---- CSR build (once; graph is shared by all 3 layers) ----
    zero_u32<<<(BN_ROWS + 255) / 256, 256, 0, stream>>>((unsigned*)deg, BN_ROWS);
    zero_u32<<<(BN_ROWS + 255) / 256, 256, 0, stream>>>((unsigned*)cursor, BN_ROWS);
    count_deg<<<(B_ * E_) / 256, 256, 0, stream>>>(edge_index, deg);
    scan_rowptr<<<1, 256, 0, stream>>>(deg, rowptr);
    fill_perm<<<(B_ * E_) / 256, 256, 0, stream>>>(edge_index, rowptr, cursor, perm);

    // h = x @ pre_w^T + pre_b   (K = D = 128)
    wmma_gemm<D_, 0><<<ggrid, 256, 0, stream>>>(x, pre_w, nullptr, nullptr,
                                                pre_b, hA);

    for (int i = 0; i < L_; ++i) {
        gather_rows<<<BN_ROWS / 8, 256, 0, stream>>>(hA, edge_index, edge_feature,
                                                     rowptr, deg, perm, agg);

        // h' = agg @ rel_w[i]^T + rel_b[i] + h @ root_w[i]^T   (K = H = 256)
        wmma_gemm<H_, 1><<<ggrid, 256, 0, stream>>>(agg, rel_w + (size_t)i * H_ * H_,
                                                    hA,  root_w + (size_t)i * H_ * H_,
                                                    rel_b + i * H_, hB);

        zero_u32<<<1, 256, 0, stream>>>((unsigned*)stats, 4 * H_);
        bn_stats<<<128, 256, 0, stream>>>(hB, stats, M / 128);
        bn_finalize<<<1, 256, 0, stream>>>(stats, bn_gamma + i * H_, bn_beta + i * H_);
        bn_apply<<<M, 256, 0, stream>>>(hB, stats);

        float* t = hA; hA = hB; hB = t;
    }

    pool_head<<<B_, 256, 0, stream>>>(hA, pool_w, pool_b, (float*)d_out);
}